// EntanglementBasedAttention_91199335563809
// MI455X (gfx1250) — compile-verified
//
#include <hip/hip_runtime.h>
#include <math.h>

#define NQ      4
#define NLAYERS 2
#define BATCH   4
#define SQ      128
#define SK      128
#define DFEAT   64
#define PI_F    3.14159265358979323846f

typedef float v2f __attribute__((ext_vector_type(2)));
typedef float v8f __attribute__((ext_vector_type(8)));

struct c32 { float re, im; };
__device__ inline c32 cmul(c32 a, c32 b) { return {a.re*b.re - a.im*b.im, a.re*b.im + a.im*b.re}; }
__device__ inline c32 cadd(c32 a, c32 b) { return {a.re + b.re, a.im + b.im}; }
__device__ inline c32 cscale(c32 a, float s) { return {a.re*s, a.im*s}; }

// One thread = one 4-qubit register simulation (16 complex amps in VGPRs).
// q registers -> SumPq (total probability, ==1 analytically, computed anyway)
// k registers -> Mk = sum_b |k_b|^2 * (1 - popcount(b)/2)  (mean <Z_i>)
__global__ void sim_registers_kernel(const float* __restrict__ query,
                                     const float* __restrict__ key,
                                     const float* __restrict__ qw,
                                     const float* __restrict__ kw,
                                     float* __restrict__ Aq,
                                     float* __restrict__ Mk)
{
    int tid = blockIdx.x * blockDim.x + threadIdx.x;
    const int NQREG = BATCH * SQ;
    const int NTOT  = NQREG + BATCH * SK;
    if (tid >= NTOT) return;

    bool isK = (tid >= NQREG);
    int  r   = isK ? (tid - NQREG) : tid;
    const float* feat = isK ? (key + (size_t)r * DFEAT) : (query + (size_t)r * DFEAT);
    const float* W    = isK ? kw : qw;

    float x[NQ];
#pragma unroll
    for (int i = 0; i < NQ; ++i) x[i] = tanhf(feat[i]);

    c32 st[16];
#pragma unroll
    for (int a = 0; a < 16; ++a) st[a] = {0.f, 0.f};
    st[0].re = 1.f;

#pragma unroll
    for (int layer = 0; layer < NLAYERS; ++layer) {
#pragma unroll
        for (int i = 0; i < NQ; ++i) {
            // RY(x_i * pi): real [[cy,-sy],[sy,cy]]
            float sy, cy; sincosf(x[i] * PI_F * 0.5f, &sy, &cy);
            // Rot(phi,theta,omega) = RZ(om) RY(th) RZ(phi)
            float phi = W[(layer*NQ + i)*3 + 0];
            float th  = W[(layer*NQ + i)*3 + 1];
            float om  = W[(layer*NQ + i)*3 + 2];
            float sth, cth; sincosf(th * 0.5f, &sth, &cth);
            float sp, cp;   sincosf(-0.5f * (phi + om), &sp, &cp);  // ep = e^{-i(phi+om)/2}
            float sm, cm;   sincosf( 0.5f * (phi - om), &sm, &cm);  // em = e^{+i(phi-om)/2}
            c32 R00 = {  cp*cth,  sp*cth };   //  ep*c
            c32 R01 = { -cm*sth, -sm*sth };   // -em*s
            c32 R10 = {  cm*sth, -sm*sth };   //  conj(em)*s
            c32 R11 = {  cp*cth, -sp*cth };   //  conj(ep)*c
            // U = Rot * RY  (RY real)
            c32 U00 = cadd(cscale(R00,  cy), cscale(R01, sy));
            c32 U01 = cadd(cscale(R00, -sy), cscale(R01, cy));
            c32 U10 = cadd(cscale(R10,  cy), cscale(R11, sy));
            c32 U11 = cadd(cscale(R10, -sy), cscale(R11, cy));
            // apply on wire i (MSB-first: bit position 3-i)
            int bit = 1 << (3 - i);
#pragma unroll
            for (int a = 0; a < 16; ++a) {
                if (a & bit) continue;
                c32 s0 = st[a], s1 = st[a | bit];
                st[a]       = cadd(cmul(U00, s0), cmul(U01, s1));
                st[a | bit] = cadd(cmul(U10, s0), cmul(U11, s1));
            }
        }
        // CNOT chain: control wire i (bit 3-i), target wire i+1 (bit 2-i)
#pragma unroll
        for (int i = 0; i < NQ - 1; ++i) {
            int cb = 1 << (3 - i), tb = 1 << (2 - i);
#pragma unroll
            for (int a = 0; a < 16; ++a) {
                if ((a & cb) && !(a & tb)) {
                    c32 t = st[a]; st[a] = st[a | tb]; st[a | tb] = t;
                }
            }
        }
    }

    if (!isK) {
        float s = 0.f;
#pragma unroll
        for (int a = 0; a < 16; ++a) s += st[a].re*st[a].re + st[a].im*st[a].im;
        Aq[r] = s;
    } else {
        float m = 0.f;
#pragma unroll
        for (int a = 0; a < 16; ++a) {
            float p2 = st[a].re*st[a].re + st[a].im*st[a].im;
            m += (1.f - 0.5f * (float)__popc((unsigned)a)) * p2;
        }
        Mk[r] = m;
    }
}

// scores[b, m, n] = 0.5 * SumPq[b,m] * Mk[b,n] + 0.5
// Rank-1 GEMM per 16x16 tile via V_WMMA_F32_16X16X4_F32 (K padded to 4).
// One wave (32 lanes) per tile; EXEC all-ones at the WMMA.
__global__ void score_wmma_kernel(const float* __restrict__ Aq,
                                  const float* __restrict__ Mk,
                                  float* __restrict__ out)
{
    int tn = blockIdx.x;          // 0..SK/16-1
    int tm = blockIdx.y;          // 0..SQ/16-1
    int b  = blockIdx.z;          // 0..BATCH-1
    int lane = threadIdx.x;       // 0..31
    int half = lane >> 4;
    int l    = lane & 15;

    // A 16x4 layout: lanes 0-15 = rows M, v0=K0,v1=K1; lanes 16-31: v0=K2,v1=K3
    // B 4x16 layout: lanes 0-15 = cols N, v0=K0,v1=K1; lanes 16-31: v0=K2,v1=K3
    v2f afrag = {0.f, 0.f};
    v2f bfrag = {0.f, 0.f};
    if (half == 0) {
        afrag.x = Aq[b * SQ + tm * 16 + l];   // A[M=l, K=0]
        bfrag.x = Mk[b * SK + tn * 16 + l];   // B[K=0, N=l]
    }
    v8f c = {0.f, 0.f, 0.f, 0.f, 0.f, 0.f, 0.f, 0.f};

    v8f d = __builtin_amdgcn_wmma_f32_16x16x4_f32(
        /*neg_a=*/false, afrag, /*neg_b=*/false, bfrag,
        /*c_mod=*/(short)0, c, /*reuse_a=*/false, /*reuse_b=*/false);

    // D layout: VGPR r -> M=r (lanes 0-15, N=lane) or M=r+8 (lanes 16-31, N=lane-16)
    int n = tn * 16 + l;
#pragma unroll
    for (int r = 0; r < 8; ++r) {
        int m = tm * 16 + r + (half ? 8 : 0);
        out[((size_t)b * SQ + m) * SK + n] = 0.5f * d[r] + 0.5f;
    }
}

extern "C" void kernel_launch(void* const* d_in, const int* in_sizes, int n_in,
                              void* d_out, int out_size, void* d_ws, size_t ws_size,
                              hipStream_t stream) {
    (void)in_sizes; (void)n_in; (void)out_size; (void)ws_size;
    const float* query = (const float*)d_in[0];
    const float* key   = (const float*)d_in[1];
    const float* qw    = (const float*)d_in[2];
    const float* kw    = (const float*)d_in[3];
    // d_in[4] (ent_weights) provably cancels out of the measured observable.

    float* Aq = (float*)d_ws;            // BATCH*SQ floats
    float* Mk = Aq + BATCH * SQ;         // BATCH*SK floats
    float* out = (float*)d_out;

    int nthreads = BATCH * (SQ + SK);    // 1024 register sims
    sim_registers_kernel<<<dim3((nthreads + 255) / 256), dim3(256), 0, stream>>>(
        query, key, qw, kw, Aq, Mk);

    score_wmma_kernel<<<dim3(SK / 16, SQ / 16, BATCH), dim3(32), 0, stream>>>(
        Aq, Mk, out);
}